// ExponentialSmoothing_21492016349900
// MI455X (gfx1250) — compile-verified
//
#include <hip/hip_runtime.h>

// Problem constants (from reference): B=8, T=4096, H=8, D=64
#define BB 8
#define TT 4096
#define HH 8
#define DD 64
#define TC 16                 // chunk length = WMMA M dimension
#define NC (TT / TC)          // 256 chunks per (b,h) series
#define ROWSTRIDE (HH * DD)   // elements between consecutive t

typedef __attribute__((ext_vector_type(2))) float v2f;
typedef __attribute__((ext_vector_type(8))) float v8f;

__device__ __forceinline__ float sigmoidf_(float x) {
    return 1.0f / (1.0f + __expf(-x));
}

// alpha^e for e >= 0, else 0.  e in [-16, 15].
// Fully unconditional square&multiply (v_mul + v_cndmask only, no branches),
// single final select -> nothing for the compiler to predicate with EXEC.
__device__ __forceinline__ float apow_or_zero(float alpha, int e) {
    const unsigned k = (unsigned)e & 15u;
    float p = alpha;
    float r = (k & 1u) ? p : 1.0f;
    p *= p;
    r = (k & 2u) ? r * p : r;
    p *= p;
    r = (k & 4u) ? r * p : r;
    p *= p;
    r = (k & 8u) ? r * p : r;
    return (e >= 0) ? r : 0.0f;
}

// u[t] coefficient for aux: factor*alpha with factor = (1-a)/max(1-a,1e-6).
// factor == 1 unless alpha ~ 1, so this is a single select (no fdiv).
__device__ __forceinline__ float aux_coeff(float alpha, float one_m) {
    return (one_m >= 1e-6f) ? alpha : one_m * 1e6f * alpha;
}

// ---------------------------------------------------------------------------
// K1: per-chunk weighted end-sum  S[b,h,c,d] = sum_{s<16} alpha^(15-s) * u[s]
// One thread per (b,h,c,d). Streams all inputs once from HBM, warms L2 for K3.
// ---------------------------------------------------------------------------
__global__ void __launch_bounds__(256) es_chunk_summary(
    const float* __restrict__ values, const float* __restrict__ aux,
    const float* __restrict__ w, float* __restrict__ S) {
    const int idx = blockIdx.x * 256 + threadIdx.x;  // [b][h][c][d] packed
    const int d = idx & (DD - 1);
    const int c = (idx >> 6) & (NC - 1);
    const int h = (idx >> 14) & (HH - 1);
    const int b = idx >> 17;

    const float alpha = sigmoidf_(w[h]);
    const float one_m = 1.0f - alpha;
    const float fa = aux_coeff(alpha, one_m);

    int off = ((b * TT + c * TC) * HH + h) * DD + d;
    float acc = 0.0f;
#pragma unroll
    for (int s = 0; s < TC; ++s) {
        __builtin_prefetch(&values[off + 4 * ROWSTRIDE], 0, 0);  // speculative
        __builtin_prefetch(&aux[off + 4 * ROWSTRIDE], 0, 0);
        const float u = fmaf(one_m, values[off], fa * aux[off]);
        acc = fmaf(alpha, acc, u);  // after loop: sum alpha^(15-s) u[s]
        off += ROWSTRIDE;
    }
    S[idx] = acc;
}

// ---------------------------------------------------------------------------
// K2: serial scan over chunk summaries (tiny): carry_in[c] = y at end of c-1
// One thread per (b,h,d); 256 dependent FMAs each.
// ---------------------------------------------------------------------------
__global__ void __launch_bounds__(256) es_carry_scan(
    const float* __restrict__ S, const float* __restrict__ v0,
    const float* __restrict__ w, float* __restrict__ carry) {
    const int idx = blockIdx.x * 256 + threadIdx.x;  // [b][h][d]
    const int d = idx & (DD - 1);
    const int h = (idx >> 6) & (HH - 1);
    const int b = idx >> 9;

    const float alpha = sigmoidf_(w[h]);
    const float a2 = alpha * alpha;
    const float a4 = a2 * a2;
    const float a8 = a4 * a4;
    const float a16 = a8 * a8;

    float cry = v0[h * DD + d];
    const int base = ((b * HH + h) * NC) * DD + d;
    for (int c = 0; c < NC; ++c) {
        carry[base + c * DD] = cry;
        cry = fmaf(a16, cry, S[base + c * DD]);
    }
}

// ---------------------------------------------------------------------------
// K3: per-chunk prefix scan as triangular matmul on the WMMA pipe.
//   Y(16t x 16d) = L(16x16) @ U(16x16),  L[i][j] = alpha^(i-j) (i>=j else 0)
// via 4x V_WMMA_F32_16X16X4_F32, then out = Y + alpha^(m+1)*carry.
// One wave per (b,h,chunk,dtile); 8 fully-active waves per block (EXEC all-1s,
// all lane-varying math is select-based, branch-free).
// Input re-reads hit the 192MB L2 (warmed by K1); output written exactly once.
// ---------------------------------------------------------------------------
__global__ void __launch_bounds__(256) es_wmma_scan(
    const float* __restrict__ values, const float* __restrict__ aux,
    const float* __restrict__ w, const float* __restrict__ carry,
    float* __restrict__ out) {
    const int wave = threadIdx.x >> 5;
    const int lane = threadIdx.x & 31;
    const int task = blockIdx.x * 8 + wave;  // dt(2) | c(8) | h(3) | b(3)
    const int dt = task & 3;
    const int c  = (task >> 2) & (NC - 1);
    const int h  = (task >> 10) & (HH - 1);
    const int b  = task >> 13;

    const int n  = lane & 15;   // N column (d within tile); also A-row M
    const int hi = lane >> 4;   // half-wave select

    const float alpha = sigmoidf_(w[h]);
    const float one_m = 1.0f - alpha;
    const float fa = aux_coeff(alpha, one_m);

    // element (t = c*16 + j, d = dt*16 + n)
    const int ebase = ((b * TT + c * TC) * HH + h) * DD + dt * 16 + n;

    v8f acc = {};
#pragma unroll
    for (int kk = 0; kk < 4; ++kk) {
        // A operand (16x4 f32): lanes 0-15 hold K = {4kk, 4kk+1},
        // lanes 16-31 hold K = {4kk+2, 4kk+3}; row M = lane&15.
        const int j0 = kk * 4 + 2 * hi;
        v2f A;
        A.x = apow_or_zero(alpha, n - j0);
        A.y = apow_or_zero(alpha, n - j0 - 1);

        // B operand (4x16 f32): VGPR v holds rows {v, v+2} split by half-wave.
        const int o0 = ebase + j0 * ROWSTRIDE;
        const int o1 = o0 + ROWSTRIDE;
        v2f Bv;
        Bv.x = fmaf(one_m, values[o0], fa * aux[o0]);
        Bv.y = fmaf(one_m, values[o1], fa * aux[o1]);

        acc = __builtin_amdgcn_wmma_f32_16x16x4_f32(
            /*neg_a=*/false, A, /*neg_b=*/false, Bv,
            /*c_mod=*/(short)0, acc, /*reuse_a=*/false, /*reuse_b=*/false);
    }

    // rank-1 cross-chunk correction + store. C/D layout: VGPR v -> M = hi*8+v.
    const float cry = carry[((b * HH + h) * NC + c) * DD + dt * 16 + n];
    const float a2 = alpha * alpha;
    const float a4 = a2 * a2;
    const float a8 = a4 * a4;
    float wp = (hi ? a8 : 1.0f) * alpha;  // alpha^(m+1) at v=0
#pragma unroll
    for (int v = 0; v < 8; ++v) {
        const int m = hi * 8 + v;
        out[ebase + m * ROWSTRIDE] = fmaf(wp, cry, acc[v]);
        wp *= alpha;
    }
}

// ---------------------------------------------------------------------------
extern "C" void kernel_launch(void* const* d_in, const int* in_sizes, int n_in,
                              void* d_out, int out_size, void* d_ws, size_t ws_size,
                              hipStream_t stream) {
    const float* values = (const float*)d_in[0];
    const float* aux    = (const float*)d_in[1];
    const float* v0     = (const float*)d_in[2];
    const float* w      = (const float*)d_in[3];
    float* out = (float*)d_out;

    // workspace: S [B,H,NC,D] then carry [B,H,NC,D]  (4 MiB each)
    float* S     = (float*)d_ws;
    float* carry = S + (size_t)BB * HH * NC * DD;

    // K1: one thread per (b,h,c,d) = 2^20 threads
    es_chunk_summary<<<(BB * HH * NC * DD) / 256, 256, 0, stream>>>(values, aux, w, S);
    // K2: one thread per (b,h,d) = 4096 threads
    es_carry_scan<<<(BB * HH * DD) / 256, 256, 0, stream>>>(S, v0, w, carry);
    // K3: one wave per (b,h,c,dtile) = 65536 waves, 8 waves/block
    es_wmma_scan<<<(BB * HH * NC * (DD / 16)) / 8, 256, 0, stream>>>(values, aux, w, carry, out);
}